// ProteinGCN_60352880443765
// MI455X (gfx1250) — compile-verified
//
#include <hip/hip_runtime.h>
#include <hip/hip_bf16.h>

// ---------------------------------------------------------------------------
// ProteinGCN on MI455X (gfx1250): fp32 WMMA GEMMs + atomic edge scatter.
// ---------------------------------------------------------------------------

typedef __attribute__((ext_vector_type(2))) float v2f;
typedef __attribute__((ext_vector_type(8))) float v8f;

#define HID_DIM 128

__device__ __forceinline__ void atom_add_f32(float* p, float v) {
  __hip_atomic_fetch_add(p, v, __ATOMIC_RELAXED, __HIP_MEMORY_SCOPE_AGENT);
}

// ---------------- degree / normalization ----------------
__global__ void k_init_deg(float* deg, int n) {
  int i = blockIdx.x * blockDim.x + threadIdx.x;
  if (i < n) deg[i] = 1.0f;  // self loop contributes 1 to every node's degree
}

__global__ void k_count_deg(const long long* __restrict__ dst, float* deg, int e) {
  int i = blockIdx.x * blockDim.x + threadIdx.x;
  if (i < e) atom_add_f32(&deg[(int)dst[i]], 1.0f);
}

__global__ void k_deg_to_dinv(float* deg, int n) {
  int i = blockIdx.x * blockDim.x + threadIdx.x;
  if (i < n) {
    float d = deg[i];
    deg[i] = (d > 0.0f) ? rsqrtf(d) : 0.0f;
  }
}

// ---------------- fp32 WMMA GEMM: C = A[MxK] @ B[KxN] (+bias)(+relu) -------
// One wave owns a 16x64 C strip (4 x 16x16 tiles sharing the A fragment).
// K consumed 4 per step with V_WMMA_F32_16X16X4_F32.
// Lane layouts per CDNA5 ISA 7.12.2:
//   A (16x4): lanes 0-15 hold M=lane, K=k+0/k+1; lanes 16-31 hold K=k+2/k+3.
//   B (4x16): vgpr0 = row k+koff, vgpr1 = row k+koff+1; N = lane&15.
//   C (16x16): acc[v] = row tm*16 + v + half*8, col = tile col + (lane&15).
// Out-of-range rows/cols are CLAMPED on load (their products land only in
// C rows/cols that are never stored) so the inner loop is branch-free and
// EXEC stays all-1s, as WMMA requires.
__global__ void __launch_bounds__(256)
k_gemm_wmma_f32(const float* __restrict__ A, const float* __restrict__ B,
                const float* __restrict__ bias, float* __restrict__ C,
                int M, int N, int K, int relu) {
  const int strips_n = (N + 63) >> 6;                 // 64-wide column strips
  const int tiles_m  = (M + 15) >> 4;
  const int strip = blockIdx.x * (blockDim.x >> 5) + (threadIdx.x >> 5);
  if (strip >= tiles_m * strips_n) return;            // wave-uniform exit
  const int tm = strip / strips_n;
  const int tn = strip - tm * strips_n;

  const int lane = threadIdx.x & 31;
  const int half = lane >> 4;       // 0: lanes 0-15, 1: lanes 16-31
  const int r    = lane & 15;
  const int koff = half * 2;        // this lane's K sub-offset within a step

  const int m  = tm * 16 + r;
  const int me = (m < M) ? m : (M - 1);               // clamped A row

  int ncol[4], nld[4];
#pragma unroll
  for (int j = 0; j < 4; ++j) {
    ncol[j] = tn * 64 + j * 16 + r;                   // C/B column, tile j
    nld[j]  = (ncol[j] < N) ? ncol[j] : (N - 1);      // clamped B column
  }

  // Flat float indices, strength-reduced in the loop (all < 2^31 here).
  int aoff = me * K + koff;
  int boff0 = koff * N + nld[0];
  int boff1 = koff * N + nld[1];
  int boff2 = koff * N + nld[2];
  int boff3 = koff * N + nld[3];
  const int bstep = 4 * N;

  v8f acc0 = {0.f,0.f,0.f,0.f,0.f,0.f,0.f,0.f};
  v8f acc1 = acc0, acc2 = acc0, acc3 = acc0;

#pragma unroll 2
  for (int k = 0; k < K; k += 4) {
    v2f a = *(const v2f*)(A + aoff);                  // 8B-aligned: K even
    aoff += 4;
    v2f b0, b1, b2, b3;
    b0.x = B[boff0]; b0.y = B[boff0 + N]; boff0 += bstep;
    b1.x = B[boff1]; b1.y = B[boff1 + N]; boff1 += bstep;
    b2.x = B[boff2]; b2.y = B[boff2 + N]; boff2 += bstep;
    b3.x = B[boff3]; b3.y = B[boff3 + N]; boff3 += bstep;
    acc0 = __builtin_amdgcn_wmma_f32_16x16x4_f32(false, a, false, b0, (short)0, acc0, false, false);
    acc1 = __builtin_amdgcn_wmma_f32_16x16x4_f32(false, a, false, b1, (short)0, acc1, false, false);
    acc2 = __builtin_amdgcn_wmma_f32_16x16x4_f32(false, a, false, b2, (short)0, acc2, false, false);
    acc3 = __builtin_amdgcn_wmma_f32_16x16x4_f32(false, a, false, b3, (short)0, acc3, false, false);
  }

  float bv[4];
#pragma unroll
  for (int j = 0; j < 4; ++j)
    bv[j] = (bias != nullptr) ? bias[nld[j]] : 0.0f;  // clamped read; unused if OOB

  v8f* accs[4] = {&acc0, &acc1, &acc2, &acc3};
#pragma unroll
  for (int v = 0; v < 8; ++v) {
    const int mm = tm * 16 + v + half * 8;
    if (mm < M) {
      float* crow = C + (long)mm * N;
#pragma unroll
      for (int j = 0; j < 4; ++j) {
        if (ncol[j] < N) {
          float val = (*accs[j])[v] + bv[j];
          if (relu) val = fmaxf(val, 0.0f);
          crow[ncol[j]] = val;
        }
      }
    }
  }
}

// ---------------- self-loop init:  agg[i,c] = h[i,c] * dinv[i]^2 ----------
__global__ void k_init_agg(const float* __restrict__ h,
                           const float* __restrict__ dinv,
                           float* __restrict__ agg, int n_nodes) {
  int i = blockIdx.x * blockDim.x + threadIdx.x;
  int node = i >> 7;  // /HID_DIM
  if (node < n_nodes) {
    float w = dinv[node];
    agg[i] = h[i] * (w * w);
  }
}

// ---------------- edge scatter: one wave per edge, 4 channels per lane ----
__global__ void k_scatter(const long long* __restrict__ src,
                          const long long* __restrict__ dst,
                          const float* __restrict__ dinv,
                          const float* __restrict__ h,
                          float* __restrict__ agg, int e) {
  int wave = (blockIdx.x * blockDim.x + threadIdx.x) >> 5;
  int lane = threadIdx.x & 31;
  if (wave >= e) return;
  int s = (int)src[wave];
  int d = (int)dst[wave];
  float w = dinv[s] * dinv[d];
  const float* hs = h + (long)s * HID_DIM;
  float* ad = agg + (long)d * HID_DIM;
#pragma unroll
  for (int j = 0; j < 4; ++j) {
    int c = lane + j * 32;
    atom_add_f32(&ad[c], hs[c] * w);
  }
}

// ---------------- bias + relu (in place) ----------------------------------
__global__ void k_bias_relu(float* __restrict__ x, const float* __restrict__ b,
                            int total) {
  int i = blockIdx.x * blockDim.x + threadIdx.x;
  if (i < total) x[i] = fmaxf(x[i] + b[i & (HID_DIM - 1)], 0.0f);
}

// ---------------- pooling --------------------------------------------------
__global__ void k_zero(float* p, int n) {
  int i = blockIdx.x * blockDim.x + threadIdx.x;
  if (i < n) p[i] = 0.0f;
}

__global__ void k_pool_accum(const float* __restrict__ h,
                             const long long* __restrict__ batch,
                             float* __restrict__ gsum, float* __restrict__ gcnt,
                             int n_nodes) {
  int wave = (blockIdx.x * blockDim.x + threadIdx.x) >> 5;
  int lane = threadIdx.x & 31;
  if (wave >= n_nodes) return;
  int g = (int)batch[wave];
  const float* hn = h + (long)wave * HID_DIM;
  float* gs = gsum + g * HID_DIM;
#pragma unroll
  for (int j = 0; j < 4; ++j) {
    int c = lane + j * 32;
    atom_add_f32(&gs[c], hn[c]);
  }
  if (lane == 0) atom_add_f32(&gcnt[g], 1.0f);
}

__global__ void k_pool_div(const float* __restrict__ gsum,
                           const float* __restrict__ gcnt,
                           float* __restrict__ pooled, int total) {
  int i = blockIdx.x * blockDim.x + threadIdx.x;
  if (i < total) {
    int g = i >> 7;  // /HID_DIM
    pooled[i] = gsum[i] / fmaxf(gcnt[g], 1.0f);
  }
}

// ---------------------------------------------------------------------------
extern "C" void kernel_launch(void* const* d_in, const int* in_sizes, int n_in,
                              void* d_out, int out_size, void* d_ws, size_t ws_size,
                              hipStream_t stream) {
  // Inputs (setup_inputs order): x, edge_index, batch, W1, b1, W2, b2, Wc, bc
  const float*     x    = (const float*)d_in[0];
  const long long* ei   = (const long long*)d_in[1];  // int64 [2, E] flat
  const long long* bat  = (const long long*)d_in[2];  // int64 [N]
  const float*     W1   = (const float*)d_in[3];
  const float*     b1   = (const float*)d_in[4];
  const float*     W2   = (const float*)d_in[5];
  const float*     b2   = (const float*)d_in[6];
  const float*     Wc   = (const float*)d_in[7];
  const float*     bc   = (const float*)d_in[8];
  float* out = (float*)d_out;

  const int IN_DIM  = 20;
  const int nNodes  = in_sizes[0] / IN_DIM;          // 100000
  const int nEdges  = in_sizes[1] / 2;               // 1600000
  const int hid     = in_sizes[4];                   // 128
  const int outDim  = in_sizes[8];                   // 30954
  const int nGraphs = out_size / outDim;             // 256

  const long long* esrc = ei;
  const long long* edst = ei + nEdges;

  // ---- workspace layout (floats, 64-float aligned) ----
  auto align64 = [](size_t v) { return (v + 63) & ~(size_t)63; };
  float* ws = (float*)d_ws;
  size_t off = 0;
  float* dinv   = ws + off; off += align64((size_t)nNodes);
  float* bufH   = ws + off; off += align64((size_t)nNodes * hid);
  float* bufA   = ws + off; off += align64((size_t)nNodes * hid);
  float* gsum   = ws + off; off += (size_t)nGraphs * hid;   // keep gcnt adjacent
  float* gcnt   = ws + off; off += align64((size_t)nGraphs);
  float* pooled = ws + off; off += align64((size_t)nGraphs * hid);
  (void)ws_size; (void)n_in;

  const int T = 256;
  const int nodeBlk  = (nNodes + T - 1) / T;
  const int edgeBlk  = (nEdges + T - 1) / T;
  const int featTot  = nNodes * hid;
  const int featBlk  = (featTot + T - 1) / T;
  const int edgeWvBlk = (nEdges + 7) / 8;   // 8 waves/block, 1 edge/wave
  const int nodeWvBlk = (nNodes + 7) / 8;

  auto gemm_blocks = [](int M, int N) {
    long strips = (long)((M + 15) / 16) * ((N + 63) / 64);
    return (int)((strips + 7) / 8);  // 8 waves (strips) per 256-thread block
  };

  // ---- normalization: deg -> dinv ----
  k_init_deg<<<nodeBlk, T, 0, stream>>>(dinv, nNodes);
  k_count_deg<<<edgeBlk, T, 0, stream>>>(edst, dinv, nEdges);
  k_deg_to_dinv<<<nodeBlk, T, 0, stream>>>(dinv, nNodes);

  // ---- layer 1: h = x @ W1 ; aggregate ; +b1, relu ----
  k_gemm_wmma_f32<<<gemm_blocks(nNodes, hid), T, 0, stream>>>(
      x, W1, nullptr, bufH, nNodes, hid, IN_DIM, 0);
  k_init_agg<<<featBlk, T, 0, stream>>>(bufH, dinv, bufA, nNodes);
  k_scatter<<<edgeWvBlk, T, 0, stream>>>(esrc, edst, dinv, bufH, bufA, nEdges);
  k_bias_relu<<<featBlk, T, 0, stream>>>(bufA, b1, featTot);

  // ---- layer 2: h = relu1 @ W2 ; aggregate ; +b2, relu ----
  k_gemm_wmma_f32<<<gemm_blocks(nNodes, hid), T, 0, stream>>>(
      bufA, W2, nullptr, bufH, nNodes, hid, hid, 0);
  k_init_agg<<<featBlk, T, 0, stream>>>(bufH, dinv, bufA, nNodes);
  k_scatter<<<edgeWvBlk, T, 0, stream>>>(esrc, edst, dinv, bufH, bufA, nEdges);
  k_bias_relu<<<featBlk, T, 0, stream>>>(bufA, b2, featTot);

  // ---- global mean pool ----
  int poolTot = nGraphs * hid;
  k_zero<<<(poolTot + nGraphs + T - 1) / T, T, 0, stream>>>(gsum, poolTot + nGraphs);
  k_pool_accum<<<nodeWvBlk, T, 0, stream>>>(bufA, bat, gsum, gcnt, nNodes);
  k_pool_div<<<(poolTot + T - 1) / T, T, 0, stream>>>(gsum, gcnt, pooled, poolTot);

  // ---- classifier: out = pooled @ Wc + bc ----
  k_gemm_wmma_f32<<<gemm_blocks(nGraphs, outDim), T, 0, stream>>>(
      pooled, Wc, bc, out, nGraphs, outDim, hid, 0);
}